// EntropyBottleneck_82557861363977
// MI455X (gfx1250) — compile-verified
//
#include <hip/hip_runtime.h>
#include <math.h>

#define CCH 192
#define BB  16
#define HW  4096                      // 64*64
#define TOTAL ((size_t)BB * CCH * HW) // 12,582,912 elements per output tensor
#define PSTRIDE 48                    // padded per-channel param stride (floats)

typedef __attribute__((ext_vector_type(4))) float v4f;   // native vector for b128 ld/st

// ---- hardware tanh (gfx1250 TRANS op) -------------------------------------
#if __has_builtin(__builtin_amdgcn_tanhf)
__device__ __forceinline__ float ftanh(float x) { return __builtin_amdgcn_tanhf(x); }
#else
__device__ __forceinline__ float ftanh(float x) {
  float r;
  // v_nop covers the documented TRANS co-execution RAW hazard (ISA 7.4 note 3)
  asm("v_tanh_f32 %0, %1\n\tv_nop" : "=v"(r) : "v"(x));
  return r;
}
#endif

__device__ __forceinline__ float softplus_f(float x) {
  // numerically stable log1p(exp(x))
  return fmaxf(x, 0.0f) + log1pf(expf(-fabsf(x)));
}

// ---- the 1->3->3->3->1 gated chain ----------------------------------------
__device__ __forceinline__ float eval_chain(float t,
    const float* A0, const float* B0v, const float* A1, const float* B1v,
    const float* A2, const float* B2v, const float* A3, float B3v,
    const float* TF) {
  float v0 = fmaf(A0[0], t, B0v[0]);
  float v1 = fmaf(A0[1], t, B0v[1]);
  float v2 = fmaf(A0[2], t, B0v[2]);
  v0 = fmaf(TF[0], ftanh(v0), v0);
  v1 = fmaf(TF[1], ftanh(v1), v1);
  v2 = fmaf(TF[2], ftanh(v2), v2);
  float w0 = fmaf(A1[0], v0, fmaf(A1[1], v1, fmaf(A1[2], v2, B1v[0])));
  float w1 = fmaf(A1[3], v0, fmaf(A1[4], v1, fmaf(A1[5], v2, B1v[1])));
  float w2 = fmaf(A1[6], v0, fmaf(A1[7], v1, fmaf(A1[8], v2, B1v[2])));
  w0 = fmaf(TF[3], ftanh(w0), w0);
  w1 = fmaf(TF[4], ftanh(w1), w1);
  w2 = fmaf(TF[5], ftanh(w2), w2);
  float u0 = fmaf(A2[0], w0, fmaf(A2[1], w1, fmaf(A2[2], w2, B2v[0])));
  float u1 = fmaf(A2[3], w0, fmaf(A2[4], w1, fmaf(A2[5], w2, B2v[1])));
  float u2 = fmaf(A2[6], w0, fmaf(A2[7], w1, fmaf(A2[8], w2, B2v[2])));
  u0 = fmaf(TF[6], ftanh(u0), u0);
  u1 = fmaf(TF[7], ftanh(u1), u1);
  u2 = fmaf(TF[8], ftanh(u2), u2);
  return fmaf(A3[0], u0, fmaf(A3[1], u1, fmaf(A3[2], u2, B3v)));
}

// ---- main: one block per (b,c) slice of 4096 elems ------------------------
// Async global->LDS double-buffered pipeline: 4 chunks of 1024 floats,
// each thread stages & consumes its private 16B per chunk (no barriers needed).
__global__ void __launch_bounds__(256)
eb_main(const float* __restrict__ x, const float* __restrict__ ws,
        float* __restrict__ out) {
  __shared__ float smem[2048];     // 2 x 4KB buffers

  const int bc = blockIdx.x;       // b*C + c  (x layout is (B,C,H,W))
  const int c  = bc % CCH;
  const float* __restrict__ P = ws + c * PSTRIDE;

  float A0[3], A1[9], A2[9], A3[3], B0v[3], B1v[3], B2v[3], TF[9];
#pragma unroll
  for (int i = 0; i < 3; ++i) A0[i]  = P[i];
#pragma unroll
  for (int i = 0; i < 9; ++i) A1[i]  = P[3 + i];
#pragma unroll
  for (int i = 0; i < 9; ++i) A2[i]  = P[12 + i];
#pragma unroll
  for (int i = 0; i < 3; ++i) A3[i]  = P[21 + i];
#pragma unroll
  for (int i = 0; i < 3; ++i) B0v[i] = P[24 + i];
#pragma unroll
  for (int i = 0; i < 3; ++i) B1v[i] = P[27 + i];
#pragma unroll
  for (int i = 0; i < 3; ++i) B2v[i] = P[30 + i];
  const float B3v = P[33];
#pragma unroll
  for (int i = 0; i < 9; ++i) TF[i]  = P[34 + i];

  const float* __restrict__ xin = x + (size_t)bc * HW;
  float* __restrict__ ox = out + (size_t)bc * HW;
  float* __restrict__ ol = out + TOTAL + (size_t)bc * HW;

  const unsigned t    = threadIdx.x;
  const unsigned lds0 = (unsigned)(size_t)&smem[0];  // LDS byte offset (flat addr[31:0])
  const unsigned long long gbase = (unsigned long long)(size_t)xin;

  // prologue: stage chunk 0
  {
    const unsigned loff = lds0 + t * 16u;
    const unsigned goff = t * 16u;
    asm volatile("global_load_async_to_lds_b128 %0, %1, %2"
                 :: "v"(loff), "v"(goff), "s"(gbase) : "memory");
  }

#pragma unroll
  for (int k = 0; k < 4; ++k) {
    if (k < 3) {
      // buffer (k+1)&1 was read >=1 iteration ago; make sure its ds_loads retired
      asm volatile("s_wait_dscnt 0x0" ::: "memory");
      const unsigned loff = lds0 + ((unsigned)((k + 1) & 1)) * 4096u + t * 16u;
      const unsigned goff = (unsigned)((k + 1) * 4096) + t * 16u;
      asm volatile("global_load_async_to_lds_b128 %0, %1, %2"
                   :: "v"(loff), "v"(goff), "s"(gbase) : "memory");
      asm volatile("s_wait_asynccnt 0x1" ::: "memory");  // chunk k landed, k+1 in flight
    } else {
      asm volatile("s_wait_asynccnt 0x0" ::: "memory");
    }

    const v4f xv = *reinterpret_cast<const v4f*>(&smem[(k & 1) * 1024 + t * 4]);
    const float in[4] = {xv.x, xv.y, xv.z, xv.w};
    float qo[4], lk[4];
#pragma unroll
    for (int j = 0; j < 4; ++j) {
      const float qv = rintf(in[j]);                    // round-half-even == jnp.round
      const float lo = eval_chain(qv - 0.5f, A0, B0v, A1, B1v, A2, B2v, A3, B3v, TF);
      const float hi = eval_chain(qv + 0.5f, A0, B0v, A1, B1v, A2, B2v, A3, B3v, TF);
      // |sigmoid(s*hi)-sigmoid(s*lo)| == 0.5*|tanh(hi/2)-tanh(lo/2)|
      const float l = 0.5f * fabsf(ftanh(0.5f * hi) - ftanh(0.5f * lo));
      qo[j] = qv;
      lk[j] = fmaxf(l, 1e-9f);
    }
    v4f q = {qo[0], qo[1], qo[2], qo[3]};
    v4f L = {lk[0], lk[1], lk[2], lk[3]};
    // outputs are write-once streams: keep them out of L2 residency (TH=NT)
    __builtin_nontemporal_store(q, reinterpret_cast<v4f*>(ox + k * 1024 + t * 4));
    __builtin_nontemporal_store(L, reinterpret_cast<v4f*>(ol + k * 1024 + t * 4));
  }
}

// ---- prep: per-channel constants -> ws ------------------------------------
// ws layout per channel (stride 48):
// [0..2]=softplus(m0) [3..11]=softplus(m1) [12..20]=softplus(m2) [21..23]=softplus(m3)
// [24..26]=b0 [27..29]=b1 [30..32]=b2 [33]=b3 [34..42]=tanh(factors[i][j]) (i*3+j)
__global__ void eb_prep(const float* __restrict__ m0, const float* __restrict__ b0,
                        const float* __restrict__ m1, const float* __restrict__ b1,
                        const float* __restrict__ m2, const float* __restrict__ b2,
                        const float* __restrict__ m3, const float* __restrict__ b3,
                        const float* __restrict__ fa, float* __restrict__ ws) {
  const int c = blockIdx.x;
  const int t = threadIdx.x;
  float r;
  if      (t <  3) r = softplus_f(m0[c * 3 + t]);
  else if (t < 12) r = softplus_f(m1[c * 9 + (t - 3)]);
  else if (t < 21) r = softplus_f(m2[c * 9 + (t - 12)]);
  else if (t < 24) r = softplus_f(m3[c * 3 + (t - 21)]);
  else if (t < 27) r = b0[c * 3 + (t - 24)];
  else if (t < 30) r = b1[c * 3 + (t - 27)];
  else if (t < 33) r = b2[c * 3 + (t - 30)];
  else if (t == 33) r = b3[c];
  else if (t < 43) {
    const int s = t - 34;
    r = ftanh(fa[((s / 3) * CCH + c) * 3 + (s % 3)]);   // factors: (3, C, 3, 1)
  } else {
    return;
  }
  ws[c * PSTRIDE + t] = r;
}

extern "C" void kernel_launch(void* const* d_in, const int* in_sizes, int n_in,
                              void* d_out, int out_size, void* d_ws, size_t ws_size,
                              hipStream_t stream) {
  // setup_inputs order: x, m0, b0, m1, b1, m2, b2, m3, b3, factors
  const float* x  = (const float*)d_in[0];
  const float* m0 = (const float*)d_in[1];
  const float* b0 = (const float*)d_in[2];
  const float* m1 = (const float*)d_in[3];
  const float* b1 = (const float*)d_in[4];
  const float* m2 = (const float*)d_in[5];
  const float* b2 = (const float*)d_in[6];
  const float* m3 = (const float*)d_in[7];
  const float* b3 = (const float*)d_in[8];
  const float* fa = (const float*)d_in[9];
  float* ws  = (float*)d_ws;    // needs C*48*4 = 36,864 bytes
  float* out = (float*)d_out;

  eb_prep<<<dim3(CCH), dim3(64), 0, stream>>>(m0, b0, m1, b1, m2, b2, m3, b3, fa, ws);
  eb_main<<<dim3(BB * CCH), dim3(256), 0, stream>>>(x, ws, out);
}